// ProbAttention_54279796687129
// MI455X (gfx1250) — compile-verified
//
#include <hip/hip_runtime.h>
#include <hip/hip_bf16.h>
#include <math.h>

// ---- problem constants (from reference setup) ----
#define PB 4
#define PH 8
#define PL 2048
#define PD 64
#define PS 40          // sample_k
#define PU 40          // top-k count
#define PUP 48         // padded to 3x16 row tiles
#define BH (PB*PH)     // 32 heads
#define SCALE 0.125f   // 1/sqrt(64)
#define NCH 16         // cumsum chunks per head
#define CHL (PL/NCH)   // 128 rows per chunk

typedef __attribute__((ext_vector_type(2))) float v2f;
typedef __attribute__((ext_vector_type(8))) float v8f;

// ------------------------------------------------------------------
// Kernel 1: sampling metric  M[bh][q] = max_s(q.k_s) - sum_s(q.k_s)/L
// one wave32 per query; lane owns 2 of the 64 dims
// ------------------------------------------------------------------
__global__ void pa_metric_kernel(const float* __restrict__ Q,
                                 const float* __restrict__ K,
                                 const int*   __restrict__ idx,
                                 float* __restrict__ M) {
    const int wave = (blockIdx.x * blockDim.x + threadIdx.x) >> 5;
    const int lane = threadIdx.x & 31;
    const int bh = wave >> 11;          // / 2048
    const int q  = wave & (PL - 1);

    const float* qv = Q + (((size_t)bh * PL) + q) * PD;
    const v2f qd = *(const v2f*)(qv + lane * 2);

    float mx = -INFINITY;
    float sm = 0.0f;
    #pragma unroll 4
    for (int s = 0; s < PS; ++s) {
        const int k = idx[q * PS + s];
        const float* kv = K + (((size_t)bh * PL) + k) * PD;
        const v2f kd = *(const v2f*)(kv + lane * 2);
        float p = qd.x * kd.x + qd.y * kd.y;
        // wave32 butterfly reduce (all lanes end with full sum)
        #pragma unroll
        for (int off = 16; off >= 1; off >>= 1)
            p += __shfl_xor(p, off, 32);
        mx = fmaxf(mx, p);
        sm += p;
    }
    if (lane == 0)
        M[(size_t)bh * PL + q] = mx - sm * (1.0f / (float)PL);
}

// ------------------------------------------------------------------
// Kernel 2: per-head top-40 of M (lowest-index tie-break, like
// jax.lax.top_k) + gather Q_top into padded [48][64] tile buffer
// one block (256 threads) per head
// ------------------------------------------------------------------
__global__ void pa_topk_gather_kernel(const float* __restrict__ M,
                                      const float* __restrict__ Q,
                                      int*   __restrict__ Mtop,
                                      float* __restrict__ Qr) {
    const int bh  = blockIdx.x;
    const int tid = threadIdx.x;

    __shared__ float vals[PL];
    __shared__ float rv[256];
    __shared__ int   ri[256];
    __shared__ int   sel[PU];

    for (int i = tid; i < PL; i += 256)
        vals[i] = M[(size_t)bh * PL + i];
    __syncthreads();

    for (int t = 0; t < PU; ++t) {
        float bv = -INFINITY;
        int   bi = PL;
        for (int i = tid; i < PL; i += 256) {
            const float v = vals[i];
            if (v > bv || (v == bv && i < bi)) { bv = v; bi = i; }
        }
        rv[tid] = bv; ri[tid] = bi;
        __syncthreads();
        for (int s = 128; s > 0; s >>= 1) {
            if (tid < s) {
                if (rv[tid + s] > rv[tid] ||
                    (rv[tid + s] == rv[tid] && ri[tid + s] < ri[tid])) {
                    rv[tid] = rv[tid + s];
                    ri[tid] = ri[tid + s];
                }
            }
            __syncthreads();
        }
        if (tid == 0) {
            sel[t] = ri[0];
            vals[ri[0]] = -INFINITY;
        }
        __syncthreads();
    }

    if (tid < PU) Mtop[bh * PU + tid] = sel[tid];

    // gather Q rows into padded 48x64 buffer (rows 40..47 = 0)
    for (int e = tid; e < PUP * PD; e += 256) {
        const int u = e >> 6;
        const int d = e & 63;
        float v = 0.0f;
        if (u < PU)
            v = Q[(((size_t)bh * PL) + sel[u]) * PD + d];
        Qr[((size_t)bh * PUP + u) * PD + d] = v;
    }
}

// ------------------------------------------------------------------
// Kernel 3: scores = Qr(48x64) @ K^T(64x2048) via f32 WMMA 16x16x4
// one wave per 16x16 output tile; scale + causal mask at store
// tiles/head: 3 row-tiles x 128 col-tiles
// ------------------------------------------------------------------
__global__ void pa_scores_wmma_kernel(const float* __restrict__ Qr,
                                      const float* __restrict__ K,
                                      const int*   __restrict__ Mtop,
                                      float* __restrict__ scores) {
    const int wave = (blockIdx.x * blockDim.x + threadIdx.x) >> 5;
    const int lane = threadIdx.x & 31;
    const int l16  = lane & 15;
    const int half = lane >> 4;          // 0: K pair 0/1, 1: K pair 2/3

    const int ct = wave & 127;           // column tile 0..127
    const int rt = (wave >> 7) % 3;      // row tile 0..2
    const int bh = wave / (128 * 3);

    const float* qbase = Qr + ((size_t)bh * PUP + (rt * 16 + l16)) * PD + 2 * half;
    const float* kbase = K  + (((size_t)bh * PL) + (ct * 16 + l16)) * PD + 2 * half;

    v8f acc = {};
    #pragma unroll
    for (int k0 = 0; k0 < PD; k0 += 4) {
        v2f a, b;
        a.x = qbase[k0];
        a.y = qbase[k0 + 1];
        b.x = kbase[k0];
        b.y = kbase[k0 + 1];
        acc = __builtin_amdgcn_wmma_f32_16x16x4_f32(
                  false, a, false, b, (short)0, acc, false, false);
    }

    const int col = ct * 16 + l16;
    #pragma unroll
    for (int v = 0; v < 8; ++v) {
        const int r = rt * 16 + v + half * 8;
        float out;
        if (r < PU) {
            const int qidx = Mtop[bh * PU + r];
            out = (col > qidx) ? -INFINITY : acc[v] * SCALE;
        } else {
            out = 0.0f;  // padded rows -> zero attn weight later
        }
        scores[((size_t)bh * PUP + r) * PL + col] = out;
    }
}

// ------------------------------------------------------------------
// Kernel 4: in-place softmax over 2048 keys; one block per (bh,u) row
// ------------------------------------------------------------------
__global__ void pa_softmax_kernel(float* __restrict__ scores) {
    const int row = blockIdx.x;          // 0..BH*PU-1
    const int bh  = row / PU;
    const int u   = row % PU;
    const int tid = threadIdx.x;
    float* p = scores + ((size_t)bh * PUP + u) * PL;

    __shared__ float red[256];

    float m = -INFINITY;
    for (int i = tid; i < PL; i += 256) m = fmaxf(m, p[i]);
    red[tid] = m;
    __syncthreads();
    for (int s = 128; s > 0; s >>= 1) {
        if (tid < s) red[tid] = fmaxf(red[tid], red[tid + s]);
        __syncthreads();
    }
    m = red[0];
    __syncthreads();

    float sum = 0.0f;
    for (int i = tid; i < PL; i += 256) {
        const float e = expf(p[i] - m);   // exp(-inf - m) == 0
        p[i] = e;
        sum += e;
    }
    red[tid] = sum;
    __syncthreads();
    for (int s = 128; s > 0; s >>= 1) {
        if (tid < s) red[tid] += red[tid + s];
        __syncthreads();
    }
    const float inv = 1.0f / red[0];
    for (int i = tid; i < PL; i += 256) p[i] *= inv;
}

// ------------------------------------------------------------------
// Kernel 5: update = attn(48x2048) @ V(2048x64) via f32 WMMA 16x16x4
// one wave per 16x16 output tile; 3 row-tiles x 4 col-tiles per head
// ------------------------------------------------------------------
__global__ void pa_update_wmma_kernel(const float* __restrict__ scores,
                                      const float* __restrict__ V,
                                      float* __restrict__ upd) {
    const int wave = (blockIdx.x * blockDim.x + threadIdx.x) >> 5;
    const int lane = threadIdx.x & 31;
    const int l16  = lane & 15;
    const int half = lane >> 4;

    const int ct = wave & 3;             // column tile 0..3
    const int rt = (wave >> 2) % 3;      // row tile 0..2
    const int bh = wave / 12;

    const float* abase = scores + ((size_t)bh * PUP + (rt * 16 + l16)) * PL + 2 * half;
    const float* vbase = V + ((size_t)bh * PL) * PD + ct * 16 + l16;

    v8f acc = {};
    for (int k0 = 0; k0 < PL; k0 += 4) {
        v2f a, b;
        a.x = abase[k0];
        a.y = abase[k0 + 1];
        b.x = vbase[(size_t)(k0 + 2 * half) * PD];
        b.y = vbase[(size_t)(k0 + 2 * half + 1) * PD];
        acc = __builtin_amdgcn_wmma_f32_16x16x4_f32(
                  false, a, false, b, (short)0, acc, false, false);
    }

    #pragma unroll
    for (int v = 0; v < 8; ++v) {
        const int r = rt * 16 + v + half * 8;
        if (r < PU)
            upd[((size_t)bh * PU + r) * PD + ct * 16 + l16] = acc[v];
    }
}

// ------------------------------------------------------------------
// Kernel 6a: per-chunk partial sums.  chunkSum[bh][c][d] = sum of
// V[bh, c*128 .. c*128+127, d].  Grid: 32*16 blocks x 64 threads.
// ------------------------------------------------------------------
__global__ void pa_chunksum_kernel(const float* __restrict__ V,
                                   float* __restrict__ csum) {
    const int blk = blockIdx.x;
    const int bh  = blk >> 4;            // / NCH
    const int c   = blk & (NCH - 1);
    const int d   = threadIdx.x;         // 0..63
    const float* src = V + (((size_t)bh * PL) + c * CHL) * PD + d;
    float s = 0.0f;
    for (int l = 0; l < CHL; ++l)
        s += src[(size_t)l * PD];
    csum[((size_t)bh * NCH + c) * PD + d] = s;
}

// ------------------------------------------------------------------
// Kernel 6b: exclusive scan of the 16 chunk sums per (bh, d).
// Grid: 32 blocks x 64 threads (trivial serial scan of 16).
// ------------------------------------------------------------------
__global__ void pa_chunkscan_kernel(const float* __restrict__ csum,
                                    float* __restrict__ cpre) {
    const int bh = blockIdx.x;
    const int d  = threadIdx.x;
    float run = 0.0f;
    #pragma unroll
    for (int c = 0; c < NCH; ++c) {
        const size_t o = ((size_t)bh * NCH + c) * PD + d;
        cpre[o] = run;
        run += csum[o];
    }
}

// ------------------------------------------------------------------
// Kernel 6c: final cumsum: acc starts at the chunk's exclusive prefix,
// then sequential scan of the 128-row chunk.  Grid: 32*16 x 64.
// ------------------------------------------------------------------
__global__ void pa_cumsum_kernel(const float* __restrict__ V,
                                 const float* __restrict__ cpre,
                                 float* __restrict__ out) {
    const int blk = blockIdx.x;
    const int bh  = blk >> 4;
    const int c   = blk & (NCH - 1);
    const int d   = threadIdx.x;
    const size_t base = (((size_t)bh * PL) + c * CHL) * PD + d;
    const float* src = V + base;
    float*       dst = out + base;
    float acc = cpre[((size_t)bh * NCH + c) * PD + d];
    for (int l = 0; l < CHL; ++l) {
        acc += src[(size_t)l * PD];
        dst[(size_t)l * PD] = acc;
    }
}

// ------------------------------------------------------------------
// Kernel 7: scatter update rows into context at M_top positions
// ------------------------------------------------------------------
__global__ void pa_scatter_kernel(const int*   __restrict__ Mtop,
                                  const float* __restrict__ upd,
                                  float* __restrict__ out) {
    const int gid = blockIdx.x * blockDim.x + threadIdx.x;
    if (gid >= BH * PU * PD) return;
    const int d  = gid & 63;
    const int u  = (gid >> 6) % PU;
    const int bh = gid / (PU * PD);
    const int row = Mtop[bh * PU + u];
    out[(((size_t)bh * PL) + row) * PD + d] = upd[((size_t)bh * PU + u) * PD + d];
}

// ------------------------------------------------------------------
extern "C" void kernel_launch(void* const* d_in, const int* in_sizes, int n_in,
                              void* d_out, int out_size, void* d_ws, size_t ws_size,
                              hipStream_t stream) {
    const float* Q   = (const float*)d_in[0];
    const float* K   = (const float*)d_in[1];
    const float* V   = (const float*)d_in[2];
    const int*   idx = (const int*)d_in[3];
    float* out = (float*)d_out;

    // workspace layout (floats)
    float* ws     = (float*)d_ws;
    float* M      = ws;                                  // BH*L           = 65536
    int*   Mtop   = (int*)(ws + (size_t)BH * PL);        // BH*U           = 1280
    float* Qr     = ws + (size_t)BH * PL + BH * PU;      // BH*48*64       = 98304
    float* scores = Qr + (size_t)BH * PUP * PD;          // BH*48*2048     = 3145728
    float* upd    = scores + (size_t)BH * PUP * PL;      // BH*40*64       = 81920
    // M is dead after kernel 2 -> reuse its 65536 floats for the scan
    float* csum   = M;                                   // BH*NCH*PD      = 32768
    float* cpre   = M + (size_t)BH * NCH * PD;           // BH*NCH*PD      = 32768
    (void)in_sizes; (void)n_in; (void)out_size; (void)ws_size;

    // 1. sampling metric: 1 wave/query -> 65536 waves -> 8192 blocks of 256
    pa_metric_kernel<<<(BH * PL) / 8, 256, 0, stream>>>(Q, K, idx, M);

    // 2. per-head top-40 + gather: 32 blocks
    pa_topk_gather_kernel<<<BH, 256, 0, stream>>>(M, Q, Mtop, Qr);

    // 3. scores WMMA: 32*3*128 = 12288 waves -> 1536 blocks of 256
    pa_scores_wmma_kernel<<<(BH * 3 * 128) / 8, 256, 0, stream>>>(Qr, K, Mtop, scores);

    // 4. softmax: one block per (bh,u) row
    pa_softmax_kernel<<<BH * PU, 256, 0, stream>>>(scores);

    // 5. update WMMA: 32*3*4 = 384 waves -> 48 blocks of 256
    pa_update_wmma_kernel<<<(BH * 3 * 4) / 8, 256, 0, stream>>>(scores, V, upd);

    // 6. hierarchical cumsum into d_out (16x the scan parallelism of a
    //    single serial pass; within-chunk order matches the reference fold)
    pa_chunksum_kernel<<<BH * NCH, PD, 0, stream>>>(V, csum);
    pa_chunkscan_kernel<<<BH, PD, 0, stream>>>(csum, cpre);
    pa_cumsum_kernel<<<BH * NCH, PD, 0, stream>>>(V, cpre, out);

    // 7. scatter selected rows
    pa_scatter_kernel<<<(BH * PU * PD + 255) / 256, 256, 0, stream>>>(Mtop, upd, out);
}